// MultiAtt_20332375179790
// MI455X (gfx1250) — compile-verified
//
#include <hip/hip_runtime.h>
#include <math.h>

// Problem constants (from reference)
#define B_TOT 262144
#define E_DIM 256

typedef __attribute__((ext_vector_type(2))) float v2f;
typedef __attribute__((ext_vector_type(8))) float v8f;

// async-to-LDS builtin operand types: v4i pointers in AS1 (global) / AS3 (LDS)
typedef __attribute__((vector_size(4 * sizeof(int)))) int v4i;
typedef __attribute__((address_space(1))) v4i gv4i;
typedef __attribute__((address_space(3))) v4i lv4i;

#if defined(__gfx1250__) && __has_builtin(__builtin_amdgcn_global_load_async_to_lds_b128)
#define HAVE_ASYNC_LDS 1
#else
#define HAVE_ASYNC_LDS 0
#endif

// ---- workspace layout (float element offsets) ----
#define OFF_W2FRAG 0u                      // 64*32*2 = 4096
#define OFF_B2     4096u                   // 16
#define OFF_GMAX   4112u                   // 9 (ordered-uint), padded 16
#define OFF_GSUM   4128u                   // 9 reciprocals, padded 16
#define OFF_SHFRAG 4144u                   // 16*3*32*2 = 3072
#define OFF_PART   7216u                   // 1024*9
#define OFF_Q      16432u                  // B*9
#define OFF_SC     (16432u + 9u * B_TOT)   // B*9 (scores -> exp in place)

__device__ __forceinline__ unsigned fenc(float f) {
  unsigned u = __float_as_uint(f);
  return (u & 0x80000000u) ? ~u : (u | 0x80000000u);
}
__device__ __forceinline__ float fdec(unsigned u) {
  unsigned v = (u & 0x80000000u) ? (u & 0x7fffffffu) : ~u;
  return __uint_as_float(v);
}

// ---------------------------------------------------------------------------
// Kernel A: fold WQ into exten_w/exten_b (W2, b2), pre-swizzle WMMA B-matrix
// fragments for both GEMMs, init reduction slots. One block of 256 threads.
// ---------------------------------------------------------------------------
__global__ void prep_kernel(const float* __restrict__ exw,
                            const float* __restrict__ exb,
                            const float* __restrict__ wq,
                            const float* __restrict__ shw,
                            float* __restrict__ ws) {
  const int tid = threadIdx.x;  // 0..255 : one k-column of W2T each

  // W2T[k][n] = sum_e WQ[s,e] * exten_w[e*3+t, k],  n = s*3+t,  k = tid
  float acc[9];
#pragma unroll
  for (int n = 0; n < 9; ++n) acc[n] = 0.f;
  for (int e = 0; e < E_DIM; ++e) {
    float w0 = wq[0 * E_DIM + e];
    float w1 = wq[1 * E_DIM + e];
    float w2 = wq[2 * E_DIM + e];
#pragma unroll
    for (int t = 0; t < 3; ++t) {
      float ew = exw[(size_t)(e * 3 + t) * E_DIM + tid];
      acc[0 * 3 + t] += w0 * ew;
      acc[1 * 3 + t] += w1 * ew;
      acc[2 * 3 + t] += w2 * ew;
    }
  }
  // Pack into WMMA f32 16x16x4 B-matrix fragment order:
  // chunk c = k/4; within chunk: VGPR elem = k&1, lane half kh = (k>>1)&1.
  {
    int c = tid >> 2, kh = (tid >> 1) & 1, elem = tid & 1;
    float* w2f = ws + OFF_W2FRAG;
#pragma unroll
    for (int n = 0; n < 16; ++n) {
      float v = (n < 9) ? acc[n] : 0.f;
      w2f[(size_t)((c * 32 + kh * 16 + n) * 2 + elem)] = v;
    }
  }

  // b2[n] = sum_e WQ[s,e] * exten_b[e*3+t]; init gmax/gsum slots.
  if (tid < 16) {
    float b = 0.f;
    if (tid < 9) {
      int s = tid / 3, t = tid % 3;
      for (int e = 0; e < E_DIM; ++e) b += wq[s * E_DIM + e] * exb[e * 3 + t];
    }
    ws[OFF_B2 + tid] = b;
    ((unsigned*)(ws + OFF_GMAX))[tid] = 0u;  // encoded -inf floor
    ws[OFF_GSUM + tid] = 0.f;
  }

  // shrink_w fragments: B[k][col] = shrink_w[col*9 + k], zero-padded K to 12.
  float* shf = ws + OFF_SHFRAG;
  for (int i = tid; i < 16 * 3 * 32; i += 256) {
    int T = i / 96, rem = i % 96, c = rem / 32, L = rem % 32;
    int n = L & 15, kh = L >> 4;
    int col = T * 16 + n;
#pragma unroll
    for (int el = 0; el < 2; ++el) {
      int kk = 4 * c + 2 * kh + el;
      shf[(size_t)i * 2 + el] = (kk < 9) ? shw[col * 9 + kk] : 0.f;
    }
  }
}

// ---------------------------------------------------------------------------
// Kernel B: Q[b][9] = x[b,:] * W2T + b2 via V_WMMA_F32_16X16X4_F32.
// One wave handles 16 batch rows. x tile staged to LDS with coalesced
// global_load_async_to_lds_b128 (ASYNCcnt), A-fragments read via ds_load_b64
// from a conflict-free padded layout. Block = 64 threads (2 waves, 33 KB LDS).
// ---------------------------------------------------------------------------
#define ROW_STRIDE 260  // 256 + 4 pad -> ds_load_b64 fragments hit 64 banks

__global__ void q_kernel(const float* __restrict__ x, float* __restrict__ ws) {
  const int lane = threadIdx.x & 31;
  const int wave = threadIdx.x >> 5;
  const int base = (blockIdx.x * 2 + wave) * 16;
  const int m = lane & 15, kh = lane >> 4;

#if HAVE_ASYNC_LDS
  __shared__ float tile[2][16 * ROW_STRIDE];
  float* my = &tile[wave][0];
  // Stage 16x256 f32 tile: 1024 b128 transfers, 32 per lane, coalesced.
#pragma unroll 4
  for (int j = 0; j < 32; ++j) {
    int i = j * 32 + lane;
    int r = i >> 6;             // row 0..15
    int c4 = (i & 63) * 4;      // float column, 16B granules
    const float* g = x + (size_t)(base + r) * E_DIM + c4;
    float* l = my + r * ROW_STRIDE + c4;
    __builtin_amdgcn_global_load_async_to_lds_b128((gv4i*)g, (lv4i*)l, 0, 0);
  }
#if __has_builtin(__builtin_amdgcn_s_wait_asynccnt)
  __builtin_amdgcn_s_wait_asynccnt(0);
#else
  asm volatile("s_wait_asynccnt 0x0" ::: "memory");
#endif
  const float* ar = my + m * ROW_STRIDE + 2 * kh;
#else
  const float* ar = x + (size_t)(base + m) * E_DIM + 2 * kh;
#endif

  const v2f* w2f = (const v2f*)(ws + OFF_W2FRAG);
  v8f acc = {0.f, 0.f, 0.f, 0.f, 0.f, 0.f, 0.f, 0.f};
#pragma unroll 4
  for (int c = 0; c < 64; ++c) {
    v2f a = *(const v2f*)(ar + 4 * c);   // A: x tile 16x4 (f32 layout)
    v2f b = w2f[c * 32 + lane];          // B: W2T chunk 4x16 (pre-swizzled)
    acc = __builtin_amdgcn_wmma_f32_16x16x4_f32(
        false, a, false, b, (short)0, acc, false, false);
  }

  const int n = lane & 15;  // output column of C tile
  if (n < 9) {
    float b2 = ws[OFF_B2 + n];
    float* Q = ws + OFF_Q;
#pragma unroll
    for (int v = 0; v < 8; ++v) {
      int row = base + v + 8 * kh;       // C layout: M = v + 8*(lane/16)
      Q[(size_t)row * 9 + n] = acc[v] + b2;
    }
  }
}

// ---------------------------------------------------------------------------
// Kernel C: scores[b] = Q Q^T / 16; deterministic global max per (s,u) slot
// via ordered-uint atomicMax (shared, then global).
// ---------------------------------------------------------------------------
__global__ void score_kernel(float* __restrict__ ws) {
  const int b = blockIdx.x * 256 + threadIdx.x;
  const float* Q = ws + OFF_Q + (size_t)b * 9;
  float q[9];
#pragma unroll
  for (int i = 0; i < 9; ++i) q[i] = Q[i];

  float sc[9];
#pragma unroll
  for (int s = 0; s < 3; ++s)
#pragma unroll
    for (int u = 0; u < 3; ++u) {
      float d = q[s * 3 + 0] * q[u * 3 + 0] + q[s * 3 + 1] * q[u * 3 + 1] +
                q[s * 3 + 2] * q[u * 3 + 2];
      sc[s * 3 + u] = d * 0.0625f;  // 1/sqrt(256)
    }

  float* SC = ws + OFF_SC + (size_t)b * 9;
#pragma unroll
  for (int i = 0; i < 9; ++i) SC[i] = sc[i];

  __shared__ unsigned smax[9];
  if (threadIdx.x < 9) smax[threadIdx.x] = 0u;
  __syncthreads();
#pragma unroll
  for (int i = 0; i < 9; ++i) atomicMax(&smax[i], fenc(sc[i]));
  __syncthreads();
  if (threadIdx.x < 9)
    atomicMax((unsigned*)(ws + OFF_GMAX) + threadIdx.x, smax[threadIdx.x]);
}

// ---------------------------------------------------------------------------
// Kernel D: e = exp(score - max) in place; deterministic per-block tree sums.
// ---------------------------------------------------------------------------
__global__ void exp_kernel(float* __restrict__ ws) {
  const int b = blockIdx.x * 256 + threadIdx.x;
  const unsigned* gm = (const unsigned*)(ws + OFF_GMAX);
  float* SC = ws + OFF_SC + (size_t)b * 9;
  float e[9];
#pragma unroll
  for (int i = 0; i < 9; ++i) {
    e[i] = __expf(SC[i] - fdec(gm[i]));
    SC[i] = e[i];
  }
  __shared__ float red[256];
  for (int su = 0; su < 9; ++su) {
    red[threadIdx.x] = e[su];
    __syncthreads();
    for (int off = 128; off > 0; off >>= 1) {
      if (threadIdx.x < off) red[threadIdx.x] += red[threadIdx.x + off];
      __syncthreads();
    }
    if (threadIdx.x == 0) ws[OFF_PART + blockIdx.x * 9 + su] = red[0];
    __syncthreads();
  }
}

// Kernel D2: reduce 1024 block partials per slot (single block, fixed order),
// store the reciprocal so the out-GEMM multiplies instead of dividing.
__global__ void sum_kernel(float* __restrict__ ws) {
  __shared__ float red[256];
  for (int su = 0; su < 9; ++su) {
    float s = 0.f;
    for (int i = threadIdx.x; i < 1024; i += 256)
      s += ws[OFF_PART + (size_t)i * 9 + su];
    red[threadIdx.x] = s;
    __syncthreads();
    for (int off = 128; off > 0; off >>= 1) {
      if (threadIdx.x < off) red[threadIdx.x] += red[threadIdx.x + off];
      __syncthreads();
    }
    if (threadIdx.x == 0) ws[OFF_GSUM + su] = 1.0f / red[0];
    __syncthreads();
  }
}

// ---------------------------------------------------------------------------
// Kernel E: att = e * (1/sum); ctx = att*Q (per-lane; lanes 16-31 mirror 0-15
// so A fragments come straight from registers); out = ctx*shrink_w^T + bias
// via WMMA. One wave: 16 batches x 256 cols (16 col-tiles x 3 K-chunks).
// ---------------------------------------------------------------------------
__global__ void out_kernel(const float* __restrict__ shb,
                           const float* __restrict__ ws,
                           float* __restrict__ out) {
  const int lane = threadIdx.x & 31;
  const int wave = threadIdx.x >> 5;
  const int base = (blockIdx.x * 8 + wave) * 16;
  const int m = lane & 15, kh = lane >> 4;
  const int b = base + m;

  const float* Q = ws + OFF_Q + (size_t)b * 9;
  const float* Ex = ws + OFF_SC + (size_t)b * 9;

  float att[9], ctx[12];
#pragma unroll
  for (int i = 0; i < 9; ++i) att[i] = Ex[i] * ws[OFF_GSUM + i];
#pragma unroll
  for (int s = 0; s < 3; ++s)
#pragma unroll
    for (int t = 0; t < 3; ++t)
      ctx[s * 3 + t] = att[s * 3 + 0] * Q[0 * 3 + t] +
                       att[s * 3 + 1] * Q[1 * 3 + t] +
                       att[s * 3 + 2] * Q[2 * 3 + t];
  ctx[9] = ctx[10] = ctx[11] = 0.f;  // pad K 9 -> 12

  v2f afr[3];
#pragma unroll
  for (int c = 0; c < 3; ++c) {
    int k = 4 * c + 2 * kh;
    afr[c].x = ctx[k];
    afr[c].y = ctx[k + 1];
  }

  const v2f* shf = (const v2f*)(ws + OFF_SHFRAG);
  const int n = lane & 15;
  for (int T = 0; T < 16; ++T) {
    v8f acc = {0.f, 0.f, 0.f, 0.f, 0.f, 0.f, 0.f, 0.f};
#pragma unroll
    for (int c = 0; c < 3; ++c) {
      acc = __builtin_amdgcn_wmma_f32_16x16x4_f32(
          false, afr[c], false, shf[(T * 3 + c) * 32 + lane], (short)0, acc,
          false, false);
    }
    int col = T * 16 + n;
    float bias = shb[col];
#pragma unroll
    for (int v = 0; v < 8; ++v) {
      int row = base + v + 8 * kh;
      out[(size_t)row * 256 + col] = acc[v] + bias;
    }
  }
}

// ---------------------------------------------------------------------------
extern "C" void kernel_launch(void* const* d_in, const int* in_sizes, int n_in,
                              void* d_out, int out_size, void* d_ws,
                              size_t ws_size, hipStream_t stream) {
  (void)in_sizes; (void)n_in; (void)out_size; (void)ws_size;
  const float* x   = (const float*)d_in[0];
  const float* exw = (const float*)d_in[1];
  const float* exb = (const float*)d_in[2];
  const float* wq  = (const float*)d_in[3];
  const float* shw = (const float*)d_in[4];
  const float* shb = (const float*)d_in[5];
  float* out = (float*)d_out;
  float* ws  = (float*)d_ws;

  prep_kernel<<<1, 256, 0, stream>>>(exw, exb, wq, shw, ws);
  q_kernel<<<B_TOT / 32, 64, 0, stream>>>(x, ws);          // 2 waves x 16 rows
  score_kernel<<<B_TOT / 256, 256, 0, stream>>>(ws);
  exp_kernel<<<B_TOT / 256, 256, 0, stream>>>(ws);
  sum_kernel<<<1, 256, 0, stream>>>(ws);
  out_kernel<<<B_TOT / 128, 256, 0, stream>>>(shb, ws, out);
}